// GlobalGatedUpdater_17085379903500
// MI455X (gfx1250) — compile-verified
//
#include <hip/hip_runtime.h>
#include <hip/hip_bf16.h>

// Native clang vector type: accepted by __builtin_nontemporal_store and lowers
// to global_load_b128 / global_store_b128 (th:TH_STORE_NT) on gfx1250.
typedef float v4f __attribute__((ext_vector_type(4)));

// Output layout: (B, ITEMS_TOTAL, DIM) float32.
// Kernel 1: out[b, :, :] = emb_table for all b. Each thread loads one float4 of
//           emb ONCE and issues `batch` independent NT B128 stores (write-BW bound;
//           16 stores outstanding on STOREcnt per thread, no intermediate waits).
// Kernel 2: for each (b, j) node entry whose index i is the LAST occurrence in
//           graph b: out[b, i, :] = (1 - alpha[i]) * emb[i, :] + alpha[i] * feat[b, j, :]

__global__ void __launch_bounds__(256)
bcast_emb_kernel(const v4f* __restrict__ emb,
                 v4f* __restrict__ out,
                 int slice_f4,   // items_total * dim / 4
                 int batch)      // number of output slices
{
    int f = blockIdx.x * 256 + threadIdx.x;
    if (f >= slice_f4) return;

    // Single read of emb (12.8 MB total across the grid -> L2/HBM read once).
    v4f v = emb[f];

    // `batch` independent non-temporal B128 stores: the 205 MB write-only stream
    // must not evict emb from L2. All stores are independent -> they pipeline on
    // STOREcnt with no waits until end-of-kernel.
    long long dst = (long long)f;
    for (int b = 0; b < batch; ++b) {
        __builtin_nontemporal_store(v, out + dst);
        dst += (long long)slice_f4;
    }
}

__global__ void __launch_bounds__(64)
apply_updates_kernel(const int*   __restrict__ nodes,      // (B*n) item indices
                     const float* __restrict__ feat,       // (B*n, dim)
                     const float* __restrict__ emb,        // (items_total, dim)
                     const float* __restrict__ alpha,      // (items_total)
                     const int*   __restrict__ n_ptr,      // device scalar n_per_graph
                     float*       __restrict__ out,        // (B, items_total, dim)
                     int items_total)
{
    __shared__ int   s_skip;
    __shared__ int   s_i;
    __shared__ int   s_b;
    __shared__ float s_a;

    const int blk = blockIdx.x;        // 0 .. B*n-1  == (b, j)
    if (threadIdx.x == 0) {
        const int n = n_ptr[0];
        const int b = blk / n;
        const int j = blk % n;
        const int i = nodes[blk];
        // last-write-wins: only the last occurrence of i within graph b writes
        int skip = 0;
        const int base = b * n;
        for (int j2 = j + 1; j2 < n; ++j2) {
            if (nodes[base + j2] == i) { skip = 1; break; }
        }
        s_skip = skip;
        s_i    = i;
        s_b    = b;
        s_a    = skip ? 0.0f : alpha[i];
    }
    __syncthreads();
    if (s_skip) return;

    const int   i   = s_i;
    const int   b   = s_b;
    const float a   = s_a;
    const int   t   = threadIdx.x;     // blockDim.x == dim
    const int   dim = blockDim.x;

    const float e  = emb[(long long)i * dim + t];
    const float fv = feat[(long long)blk * dim + t];
    // match reference rounding: (1 - alpha) * emb + alpha * feat
    const float r  = (1.0f - a) * e + a * fv;
    out[((long long)b * items_total + (long long)i) * (long long)dim + t] = r;
}

extern "C" void kernel_launch(void* const* d_in, const int* in_sizes, int n_in,
                              void* d_out, int out_size, void* d_ws, size_t ws_size,
                              hipStream_t stream) {
    const int*   nodes = (const int*)  d_in[0];   // (B*n) indices
    const float* feat  = (const float*)d_in[1];   // (B*n, dim)
    const float* emb   = (const float*)d_in[2];   // (items_total, dim)
    const float* alpha = (const float*)d_in[3];   // (items_total, 1)
    const int*   n_ptr = (const int*)  d_in[4];   // scalar n_per_graph (device)
    float*       out   = (float*)d_out;

    const int n_nodes     = in_sizes[0];                    // B * n  (== 800)
    const int items_total = in_sizes[3];                    // 50000
    const int dim         = in_sizes[2] / items_total;      // 64
    const long long slice = (long long)items_total * dim;   // elems per batch slice
    const int batch       = (int)((long long)out_size / slice);  // 16
    const int slice_f4    = (int)(slice / 4);               // dim is a multiple of 4

    // Kernel 1: broadcast copy emb -> all batch slices (store-BW bound).
    dim3 gridA((slice_f4 + 255) / 256, 1, 1);
    bcast_emb_kernel<<<gridA, 256, 0, stream>>>(
        (const v4f*)emb, (v4f*)out, slice_f4, batch);

    // Kernel 2: overwrite the (<= B*n) gated rows. Same stream -> ordered after copy.
    apply_updates_kernel<<<n_nodes, dim, 0, stream>>>(
        nodes, feat, emb, alpha, n_ptr, out, items_total);
}